// calculate_score_25262997635142
// MI455X (gfx1250) — compile-verified
//
#include <hip/hip_runtime.h>
#include <stdint.h>

#define T_TOTAL 500000
#define STRINGS 6
#define FRETS   21
#define PITCHES 45
#define CHUNK   64
#define NTHREADS 256

// d_ws counter layout (unsigned int):
// 0: tab_TP, 1: tab_P(pred positives), 2: tab_G(gt positives)
// 3: f0_TP,  4: f0_P,                  5: f0_G

__global__ void init_counters_kernel(unsigned int* c) {
    if (threadIdx.x < 6) c[threadIdx.x] = 0u;
}

__global__ __launch_bounds__(NTHREADS)
void score_count_kernel(const float* __restrict__ tab_pred,
                        const float* __restrict__ F0_pred,
                        const float* __restrict__ tab_gt,
                        const float* __restrict__ F0_gt,
                        const int* __restrict__ gt_len_p,
                        unsigned int* __restrict__ counters)
{
    __shared__ __align__(16) float s_tab[CHUNK * STRINGS * FRETS]; // 32256 B
    __shared__ __align__(16) float s_f0[CHUNK * PITCHES];          // 11520 B
    __shared__ unsigned int s_cnt[6];

    const int gt_len = *gt_len_p;
    const int t0  = blockIdx.x * CHUNK;
    const int len = min(CHUNK, T_TOTAL - t0);

    // ---- Stage tab_pred chunk into LDS via CDNA5 async load-to-LDS ----
    {
        const float* src = tab_pred + (size_t)t0 * (STRINGS * FRETS);
        const int nf = len * STRINGS * FRETS;
        const int nv = nf >> 2;                       // 16B packets
        for (int i = threadIdx.x; i < nv; i += NTHREADS) {
            unsigned           lo = (unsigned)(uintptr_t)(&s_tab[i * 4]);
            unsigned long long ga = (unsigned long long)(uintptr_t)(src + i * 4);
            asm volatile("global_load_async_to_lds_b128 %0, %1, off"
                         :: "v"(lo), "v"(ga) : "memory");
        }
        for (int i = (nv << 2) + threadIdx.x; i < nf; i += NTHREADS) { // tail (safety)
            unsigned           lo = (unsigned)(uintptr_t)(&s_tab[i]);
            unsigned long long ga = (unsigned long long)(uintptr_t)(src + i);
            asm volatile("global_load_async_to_lds_b32 %0, %1, off"
                         :: "v"(lo), "v"(ga) : "memory");
        }
    }
    // ---- Stage F0_pred chunk ----
    {
        const float* src = F0_pred + (size_t)t0 * PITCHES;
        const int nf = len * PITCHES;
        const int nv = nf >> 2;
        for (int i = threadIdx.x; i < nv; i += NTHREADS) {
            unsigned           lo = (unsigned)(uintptr_t)(&s_f0[i * 4]);
            unsigned long long ga = (unsigned long long)(uintptr_t)(src + i * 4);
            asm volatile("global_load_async_to_lds_b128 %0, %1, off"
                         :: "v"(lo), "v"(ga) : "memory");
        }
        for (int i = (nv << 2) + threadIdx.x; i < nf; i += NTHREADS) {
            unsigned           lo = (unsigned)(uintptr_t)(&s_f0[i]);
            unsigned long long ga = (unsigned long long)(uintptr_t)(src + i);
            asm volatile("global_load_async_to_lds_b32 %0, %1, off"
                         :: "v"(lo), "v"(ga) : "memory");
        }
    }
    asm volatile("s_wait_asynccnt 0x0" ::: "memory");
    __syncthreads();

    unsigned tabTP = 0, tabP = 0, tabG = 0;
    unsigned f0TP = 0, f0P = 0, f0G = 0;

    // ---- tab: one (t, string) row per thread iteration ----
    const int rows = len * STRINGS;
    for (int r = threadIdx.x; r < rows; r += NTHREADS) {
        const int t = t0 + r / STRINGS;
        const float* row = &s_tab[r * FRETS];          // stride 21 dwords: bank-conflict free
        int p = 0; float best = row[0];
        #pragma unroll
        for (int k = 1; k < FRETS; ++k) {
            float v = row[k];
            if (v > best) { best = v; p = k; }         // first-max tie-break, matches argmax
        }
        const float* grow = tab_gt + ((size_t)t0 * STRINGS + r) * FRETS;
        const bool masked = t < gt_len;
        const bool predpos = masked && (p < FRETS - 1);
        if (predpos)                 tabP++;
        if (predpos && grow[p] != 0.0f) tabTP++;
        if (grow[FRETS - 1] == 0.0f)    tabG++;
    }

    // ---- F0: one timestep per thread iteration ----
    for (int tl = threadIdx.x; tl < len; tl += NTHREADS) {
        const int t = t0 + tl;
        const float* row = &s_f0[tl * PITCHES];        // stride 45 dwords: bank-conflict free
        int p = 0; float best = row[0];
        #pragma unroll
        for (int k = 1; k < PITCHES; ++k) {
            float v = row[k];
            if (v > best) { best = v; p = k; }
        }
        const float* grow = F0_gt + (size_t)t * PITCHES;
        const bool masked = t < gt_len;
        const bool predpos = masked && (p < PITCHES - 1);
        if (predpos)                 f0P++;
        if (predpos && grow[p] != 0.0f) f0TP++;
        if (grow[PITCHES - 1] == 0.0f)  f0G++;
    }

    // ---- block reduction, then 6 global atomics ----
    if (threadIdx.x < 6) s_cnt[threadIdx.x] = 0u;
    __syncthreads();
    atomicAdd(&s_cnt[0], tabTP);
    atomicAdd(&s_cnt[1], tabP);
    atomicAdd(&s_cnt[2], tabG);
    atomicAdd(&s_cnt[3], f0TP);
    atomicAdd(&s_cnt[4], f0P);
    atomicAdd(&s_cnt[5], f0G);
    __syncthreads();
    if (threadIdx.x < 6) atomicAdd(&counters[threadIdx.x], s_cnt[threadIdx.x]);
}

__device__ inline void write_scores(float* out, long long TP, long long P,
                                    long long G, long long total)
{
    const long long FP = P - TP;
    const long long FN = G - TP;
    const long long TN = total - TP - FP - FN;
    const float fTP = (float)TP, fTN = (float)TN, fFP = (float)FP, fFN = (float)FN;
    const float dp = fTP + fFP;
    const float dr = fTP + fFN;
    const float prec = (dp > 0.0f) ? fTP / dp : 0.0f;   // nan_to_num(0/0) -> 0
    const float rec  = (dr > 0.0f) ? fTP / dr : 0.0f;
    const float ds   = prec + rec;
    const float f1   = (ds > 0.0f) ? 2.0f * prec * rec / ds : 0.0f;
    const float acc  = (fTP + fTN) / (fTP + fFN + fTN + fFP);
    out[0] = prec; out[1] = rec; out[2] = f1; out[3] = acc;
}

__global__ void finalize_kernel(const unsigned int* __restrict__ c,
                                float* __restrict__ out)
{
    if (threadIdx.x != 0 || blockIdx.x != 0) return;
    const long long tab_total = (long long)T_TOTAL * STRINGS * (FRETS - 1);
    const long long f0_total  = (long long)T_TOTAL * (PITCHES - 1);
    write_scores(out + 0, (long long)c[0], (long long)c[1], (long long)c[2], tab_total);
    write_scores(out + 4, (long long)c[3], (long long)c[4], (long long)c[5], f0_total);
}

extern "C" void kernel_launch(void* const* d_in, const int* in_sizes, int n_in,
                              void* d_out, int out_size, void* d_ws, size_t ws_size,
                              hipStream_t stream)
{
    (void)in_sizes; (void)n_in; (void)out_size; (void)ws_size;
    const float* tab_pred = (const float*)d_in[0];
    const float* F0_pred  = (const float*)d_in[1];
    const float* tab_gt   = (const float*)d_in[2];
    const float* F0_gt    = (const float*)d_in[3];
    const int*   gt_len   = (const int*)d_in[4];
    float* out = (float*)d_out;
    unsigned int* counters = (unsigned int*)d_ws;

    init_counters_kernel<<<1, 32, 0, stream>>>(counters);

    const int nblocks = (T_TOTAL + CHUNK - 1) / CHUNK;   // 7813
    score_count_kernel<<<nblocks, NTHREADS, 0, stream>>>(
        tab_pred, F0_pred, tab_gt, F0_gt, gt_len, counters);

    finalize_kernel<<<1, 32, 0, stream>>>(counters, out);
}